// Attention_new_77738908057670
// MI455X (gfx1250) — compile-verified
//
#include <hip/hip_runtime.h>
#include <hip/hip_bf16.h>
#include <math.h>

// MI455X (gfx1250): wave32, V_WMMA_F32_16X16X4_F32 for all GEMMs,
// GLOBAL_LOAD_ASYNC_TO_LDS_B128 (ASYNCcnt) for LDS tile staging.
// Shapes: B=8, N=256, DIM=128, H=8, d=16.

typedef __attribute__((ext_vector_type(2))) float v2f;
typedef __attribute__((ext_vector_type(8))) float v8f;

#define NB   8
#define NN   256
#define CDIM 128
#define NH   8
#define DD   16
#define BH   (NB * NH)            // 64
#define INV_SQRT_SCALE 3.3635856610148585f   // 128^(1/4)

static __device__ __forceinline__ v8f wmma_f32_4(v2f a, v2f b, v8f c) {
  return __builtin_amdgcn_wmma_f32_16x16x4_f32(false, a, false, b, (short)0, c,
                                               false, false);
}

// Async copy 16 bytes global -> LDS (tracked on ASYNCcnt).  GV mode: 64-bit
// address in VGPR pair, saddr=off.  VDST VGPR holds the LDS byte address
// (low 32 bits of the generic pointer == LDS offset per the aperture map).
static __device__ __forceinline__ void async_copy_b128(const float* g,
                                                       float* l) {
  unsigned loff = (unsigned)(size_t)l;
  asm volatile("global_load_async_to_lds_b128 %0, %1, off" ::"v"(loff), "v"(g)
               : "memory");
}
static __device__ __forceinline__ void async_wait0() {
  asm volatile("s_wait_asynccnt 0x0" ::: "memory");
}

// ---------------------------------------------------------------------------
// Generic Y[R,128] = X[R,128] @ W[128,128] + bias.  Grid.x = R/16, 256 thr.
// X tile (16x128, 8 KB) staged once via async-to-LDS; wave w owns output
// cols [16w,16w+16); 32 WMMA per wave (K=128).  W/bias stay in global
// (64 KB, L0/L2 resident across thousands of blocks).
// ---------------------------------------------------------------------------
__global__ __launch_bounds__(256) void linear128_wmma(
    const float* __restrict__ X, const float* __restrict__ W,
    const float* __restrict__ bias, float* __restrict__ Y) {
  const int row0 = blockIdx.x << 4;
  __shared__ float Xs[16 * CDIM];  // 8 KB
  {
    const int t4 = threadIdx.x << 2;  // 256 thr * 4 floats = 1024 per shot
    const float* src = X + (size_t)row0 * CDIM;
    async_copy_b128(src + t4, Xs + t4);
    async_copy_b128(src + t4 + 1024, Xs + t4 + 1024);
    async_wait0();
  }
  __syncthreads();

  const int wv   = threadIdx.x >> 5;
  const int lane = threadIdx.x & 31;
  const int half = lane >> 4;
  const int ln   = lane & 15;
  const int n0   = wv << 4;

  const float* Xr = Xs + ln * CDIM;
  v8f c = {};
#pragma unroll
  for (int kk = 0; kk < CDIM; kk += 4) {
    v2f a, b;
    a.x = Xr[kk + 2 * half];
    a.y = Xr[kk + 2 * half + 1];
    const float* bp = W + (size_t)(kk + 2 * half) * CDIM + n0 + ln;
    b.x = bp[0];
    b.y = bp[CDIM];
    c = wmma_f32_4(a, b, c);
  }
  const float bb = bias[n0 + ln];
  float* Yb = Y + (size_t)row0 * CDIM + n0 + ln;
#pragma unroll
  for (int r = 0; r < 8; ++r) Yb[(size_t)(r + 8 * half) * CDIM] = c[r] + bb;
}

// ---------------------------------------------------------------------------
// S[bh,m,n] = 128^(1/4) * sum_dd Q[bh,m,dd] * K[bh,m,n,dd]
// ---------------------------------------------------------------------------
__global__ __launch_bounds__(256) void scores_kernel(
    const float* __restrict__ qmat, const float* __restrict__ kmat,
    float* __restrict__ S) {
  const int m  = blockIdx.x & (NN - 1);
  const int bh = blockIdx.x >> 8;
  const int n  = threadIdx.x;
  __shared__ float qs[DD];
  if (threadIdx.x < DD)
    qs[threadIdx.x] = qmat[(size_t)bh * 4096 + (size_t)m * DD + threadIdx.x];
  __syncthreads();
  const float* kp =
      kmat + (size_t)bh * 1048576 + (size_t)m * 4096 + (size_t)n * DD;
  float s = 0.f;
#pragma unroll
  for (int d = 0; d < DD; ++d) s += kp[d] * qs[d];
  S[(size_t)bh * 65536 + (size_t)m * NN + n] = s * INV_SQRT_SCALE;
}

// Row softmax stats (over n).  Block = (bh,m).
__global__ __launch_bounds__(256) void rowstats_kernel(
    const float* __restrict__ S, float* __restrict__ rmax,
    float* __restrict__ rsum) {
  const int idx = blockIdx.x;  // bh*256 + m
  const int t   = threadIdx.x;
  const float v = S[(size_t)idx * NN + t];
  __shared__ float red[NN];
  red[t] = v;
  __syncthreads();
  for (int s = NN / 2; s > 0; s >>= 1) {
    if (t < s) red[t] = fmaxf(red[t], red[t + s]);
    __syncthreads();
  }
  const float mx = red[0];
  __syncthreads();
  red[t] = __expf(v - mx);
  __syncthreads();
  for (int s = NN / 2; s > 0; s >>= 1) {
    if (t < s) red[t] += red[t + s];
    __syncthreads();
  }
  if (t == 0) {
    rmax[idx] = mx;
    rsum[idx] = red[0];
  }
}

// Column softmax stats (over m).  Block = (bh,n).
__global__ __launch_bounds__(256) void colstats_kernel(
    const float* __restrict__ S, float* __restrict__ cmax,
    float* __restrict__ csum) {
  const int n  = blockIdx.x & (NN - 1);
  const int bh = blockIdx.x >> 8;
  const int t  = threadIdx.x;
  const float v = S[(size_t)bh * 65536 + (size_t)t * NN + n];
  __shared__ float red[NN];
  red[t] = v;
  __syncthreads();
  for (int s = NN / 2; s > 0; s >>= 1) {
    if (t < s) red[t] = fmaxf(red[t], red[t + s]);
    __syncthreads();
  }
  const float mx = red[0];
  __syncthreads();
  red[t] = __expf(v - mx);
  __syncthreads();
  for (int s = NN / 2; s > 0; s >>= 1) {
    if (t < s) red[t] += red[t + s];
    __syncthreads();
  }
  if (t == 0) {
    cmax[bh * NN + n] = mx;
    csum[bh * NN + n] = red[0];
  }
}

// message[m,n] = out_attn[m,n] + in_attn[m,n] - (m==n ? out_attn[m,m] : 0)
//   out_attn[m,n] = exp(S[m,n]-rmax[m])/rsum[m]
//   in_attn [m,n] = exp(S[n,m]-cmax[m])/csum[m]   (column-m softmax of S)
__global__ __launch_bounds__(256) void message_kernel(
    const float* __restrict__ S, const float* __restrict__ rmax,
    const float* __restrict__ rsum, const float* __restrict__ cmax,
    const float* __restrict__ csum, float* __restrict__ MSG) {
  const int m  = blockIdx.x & (NN - 1);
  const int bh = blockIdx.x >> 8;
  const int n  = threadIdx.x;
  const size_t base = (size_t)bh * 65536;
  const float smn = S[base + (size_t)m * NN + n];
  const float snm = S[base + (size_t)n * NN + m];
  const float outp = __expf(smn - rmax[bh * NN + m]) / rsum[bh * NN + m];
  const float inp  = __expf(snm - cmax[bh * NN + m]) / csum[bh * NN + m];
  MSG[base + (size_t)m * NN + n] = (m == n) ? inp : (outp + inp);
}

// ---------------------------------------------------------------------------
// NH[bh] = MSG[bh] (256x256) @ V[bh] (256x16).  One block per bh, 16 waves,
// each wave one 16-row tile; 64 WMMA per wave.  V staged via async-to-LDS.
// ---------------------------------------------------------------------------
__global__ __launch_bounds__(512) void nh_kernel(
    const float* __restrict__ MSG, const float* __restrict__ vmat,
    float* __restrict__ NHp) {
  const int bh = blockIdx.x;
  __shared__ float Vs[NN * DD];  // 16 KB
  {
    const int t4 = threadIdx.x << 2;  // 512 thr * 4 floats = 2048 per shot
    const float* src = vmat + (size_t)bh * 4096;
    async_copy_b128(src + t4, Vs + t4);
    async_copy_b128(src + t4 + 2048, Vs + t4 + 2048);
    async_wait0();
  }
  __syncthreads();
  const int wv   = threadIdx.x >> 5;  // m-tile 0..15
  const int lane = threadIdx.x & 31;
  const int half = lane >> 4;
  const int ln   = lane & 15;
  const float* Mr = MSG + (size_t)bh * 65536 + (size_t)(wv * 16 + ln) * NN;
  v8f c = {};
  for (int kk = 0; kk < NN; kk += 4) {
    v2f a, b;
    a.x = Mr[kk + 2 * half];
    a.y = Mr[kk + 2 * half + 1];
    b.x = Vs[(kk + 2 * half) * DD + ln];
    b.y = Vs[(kk + 2 * half + 1) * DD + ln];
    c = wmma_f32_4(a, b, c);
  }
  float* Np = NHp + (size_t)bh * 4096 + (size_t)(wv * 16) * DD + ln;
#pragma unroll
  for (int r = 0; r < 8; ++r) Np[(r + 8 * half) * DD] = c[r];
}

// ---------------------------------------------------------------------------
// Per (bh, a):
//   stage1: E (256x16) = MSG[bh] @ K[bh,a]   (K_a async-staged in LDS)
//   stage2: view E as T (32x128); out tile = T @ We + be -> edge_hidden
// One block per (bh,a), 16 waves. 64 + 32 WMMA per wave.
// ---------------------------------------------------------------------------
__global__ __launch_bounds__(512) void edge_kernel(
    const float* __restrict__ MSG, const float* __restrict__ kmat,
    const float* __restrict__ We, const float* __restrict__ be,
    float* __restrict__ EH) {
  const int a  = blockIdx.x & (NN - 1);
  const int bh = blockIdx.x >> 8;
  __shared__ float Ks[NN * DD];  // 16 KB
  __shared__ float Es[NN * DD];  // 16 KB, doubles as T[32][128]
  {
    const int t4 = threadIdx.x << 2;
    const float* src = kmat + (size_t)bh * 1048576 + (size_t)a * 4096;
    async_copy_b128(src + t4, Ks + t4);
    async_copy_b128(src + t4 + 2048, Ks + t4 + 2048);
    async_wait0();
  }
  __syncthreads();

  const int wv   = threadIdx.x >> 5;
  const int lane = threadIdx.x & 31;
  const int half = lane >> 4;
  const int ln   = lane & 15;

  {  // stage 1: E = M @ K_a
    const float* Mr = MSG + (size_t)bh * 65536 + (size_t)(wv * 16 + ln) * NN;
    __builtin_prefetch(Mr, 0, 1);  // global_prefetch_b8 (row is L2-hot)
    v8f c = {};
    for (int kk = 0; kk < NN; kk += 4) {
      v2f aa, bb;
      aa.x = Mr[kk + 2 * half];
      aa.y = Mr[kk + 2 * half + 1];
      bb.x = Ks[(kk + 2 * half) * DD + ln];
      bb.y = Ks[(kk + 2 * half + 1) * DD + ln];
      c = wmma_f32_4(aa, bb, c);
    }
#pragma unroll
    for (int r = 0; r < 8; ++r) Es[(wv * 16 + r + 8 * half) * DD + ln] = c[r];
  }
  __syncthreads();

  {  // stage 2: (E viewed as 32x128) @ We + be
    const int rt = wv >> 3;  // 0..1  : 16-row tile of T
    const int ct = wv & 7;   // 0..7  : 16-col tile of output
    const float* Ta = Es + (size_t)(rt * 16 + ln) * CDIM;  // flat row of T
    v8f c = {};
#pragma unroll
    for (int kk = 0; kk < CDIM; kk += 4) {
      v2f aa, bb;
      aa.x = Ta[kk + 2 * half];
      aa.y = Ta[kk + 2 * half + 1];
      const float* bp = We + (size_t)(kk + 2 * half) * CDIM + ct * 16 + ln;
      bb.x = bp[0];
      bb.y = bp[CDIM];
      c = wmma_f32_4(aa, bb, c);
    }
    const int hh = bh & (NH - 1);
    const int b  = bh >> 3;
    const int i  = hh * 32 + (a >> 3);
    const int j0 = (a & 7) * 32;
    const float bias = be[ct * 16 + ln];
#pragma unroll
    for (int r = 0; r < 8; ++r) {
      const int jrow = rt * 16 + r + 8 * half;
      EH[(((size_t)(b * NN + i)) * NN + (j0 + jrow)) * CDIM + ct * 16 + ln] =
          c[r] + bias;
    }
  }
}

// ---------------------------------------------------------------------------
extern "C" void kernel_launch(void* const* d_in, const int* in_sizes, int n_in,
                              void* d_out, int out_size, void* d_ws,
                              size_t ws_size, hipStream_t stream) {
  const float* node = (const float*)d_in[0];
  const float* edge = (const float*)d_in[1];
  const float* Wq   = (const float*)d_in[2];
  const float* bq   = (const float*)d_in[3];
  const float* Wk   = (const float*)d_in[4];
  const float* bk   = (const float*)d_in[5];
  const float* Wv   = (const float*)d_in[6];
  const float* bv   = (const float*)d_in[7];
  const float* Wn   = (const float*)d_in[8];
  const float* bn   = (const float*)d_in[9];
  const float* We   = (const float*)d_in[10];
  const float* be   = (const float*)d_in[11];

  float* out   = (float*)d_out;
  float* nodeH = out;                          // [8,256,128]       = 262144
  float* edgeH = out + 262144;                 // [8,256,256,128]   = 67108864
  float* MSG   = out + 262144 + 67108864;      // [8,8,256,256]     = 4194304

  float* ws   = (float*)d_ws;
  float* qmat = ws;                  // 262144
  float* vmat = qmat + 262144;       // 262144
  float* kmat = vmat + 262144;       // 67108864
  float* S    = kmat + 67108864;     // 4194304
  float* rmax = S + 4194304;         // 16384
  float* rsum = rmax + 16384;        // 16384
  float* cmax = rsum + 16384;        // 16384
  float* csum = cmax + 16384;        // 16384
  float* nhp  = csum + 16384;        // 262144

  // Linear embeddings (WMMA f32 16x16x4)
  linear128_wmma<<<128, 256, 0, stream>>>(node, Wq, bq, qmat);          // q
  linear128_wmma<<<128, 256, 0, stream>>>(node, Wv, bv, vmat);          // v
  linear128_wmma<<<32768, 256, 0, stream>>>(edge, Wk, bk, kmat);        // k

  // Scores + dual softmax + message (written directly into d_out)
  scores_kernel<<<BH * NN, 256, 0, stream>>>(qmat, kmat, S);
  rowstats_kernel<<<BH * NN, 256, 0, stream>>>(S, rmax, rsum);
  colstats_kernel<<<BH * NN, 256, 0, stream>>>(S, cmax, csum);
  message_kernel<<<BH * NN, 256, 0, stream>>>(S, rmax, rsum, cmax, csum, MSG);

  // node_hidden = (message @ v_embed) @ Wn + bn
  nh_kernel<<<BH, 512, 0, stream>>>(MSG, vmat, nhp);
  linear128_wmma<<<128, 256, 0, stream>>>(nhp, Wn, bn, nodeH);

  // edge_hidden = (message @ k_embed) @ We + be, fused per (bh,a) block
  edge_kernel<<<BH * NN, 512, 0, stream>>>(MSG, kmat, We, be, edgeH);
}